// GlobalPointer_1322849927839
// MI455X (gfx1250) — compile-verified
//
#include <hip/hip_runtime.h>
#include <math.h>

typedef __attribute__((ext_vector_type(2))) float v2f;
typedef __attribute__((ext_vector_type(8))) float v8f;

#define NEG_BIG 1000000000000.0f
// log2(10000) / 32  (RoPE inverse-frequency exponent step, base-2)
#define ROPE_L2_STEP 0.4152410118609203f

// ---------------------------------------------------------------------------
// Kernel 1: projection GEMM (4096x768 @ 768x2048) + bias + fused RoPE.
// One wave32 per 32x16 output tile: two 16x16 accumulators (M and M+16)
// sharing the B/W fragment -> 2 independent WMMA chains per K-step.
// A fragment: lane l (l15=l&15) holds row tm+l15, K pair (kh*2, kh*2+1).
// B fragment: lane l holds col tn+l15, K pair (kh*2, kh*2+1).
// D layout:   vgpr r, lanes 0-15 -> M=tm+r, lanes 16-31 -> M=tm+8+r; N=tn+l15.
// RoPE mixes columns (2i, 2i+1) which live in adjacent lanes -> shfl_xor(1).
// ---------------------------------------------------------------------------
__global__ __launch_bounds__(256) void gp_gemm_rope(
    const float* __restrict__ X, const float* __restrict__ W,
    const float* __restrict__ bias, float* __restrict__ qk)
{
  const int wave = blockIdx.x * 8 + ((int)threadIdx.x >> 5);
  const int lane = (int)threadIdx.x & 31;
  const int l15  = lane & 15;
  const int kh   = lane >> 4;            // 0 -> K {0,1}, 1 -> K {2,3}
  const int tn   = (wave & 127) << 4;    // N tile base, N = 2048
  const int tm0  = (wave >> 7) << 5;     // M pair base, M = 4096
  const int tm1  = tm0 + 16;

  const float* Arow0 = X + (size_t)(tm0 + l15) * 768 + (kh << 1);
  const float* Arow1 = Arow0 + (size_t)16 * 768;
  const float* Bcol  = W + (size_t)(kh << 1) * 2048 + tn + l15;

  v8f acc0 = {};
  v8f acc1 = {};
#pragma unroll 8
  for (int k0 = 0; k0 < 768; k0 += 4) {
    v2f a0 = *(const v2f*)(Arow0 + k0);        // aligned float2
    v2f a1 = *(const v2f*)(Arow1 + k0);
    v2f b;
    b.x = Bcol[(size_t)k0 * 2048];             // W[k0+kh*2    ][n]
    b.y = Bcol[(size_t)k0 * 2048 + 2048];      // W[k0+kh*2 + 1][n]
    acc0 = __builtin_amdgcn_wmma_f32_16x16x4_f32(false, a0, false, b,
                                                 (short)0, acc0, false, false);
    acc1 = __builtin_amdgcn_wmma_f32_16x16x4_f32(false, a1, false, b,
                                                 (short)0, acc1, false, false);
  }

  // Epilogue: bias + RoPE, store to workspace qk[4096][2048].
  const int   col  = tn + l15;
  const float bcol = bias[col];
  const int   i2   = (col >> 1) & 31;          // rope frequency index (0..31)
  const float inv  = exp2f(-(float)i2 * ROPE_L2_STEP);  // 10000^(-2i/64)
  const int   odd  = col & 1;
#pragma unroll
  for (int r = 0; r < 8; ++r) {
    {
      const int   m  = tm0 + r + (kh << 3);
      const float v  = acc0[r] + bcol;
      const float vp = __shfl_xor(v, 1, 32);   // partner column (col ^ 1)
      float sn, cs;
      __sincosf((float)(m & 511) * inv, &sn, &cs);
      qk[(size_t)m * 2048 + col] = odd ? (v * cs + vp * sn) : (v * cs - vp * sn);
    }
    {
      const int   m  = tm1 + r + (kh << 3);
      const float v  = acc1[r] + bcol;
      const float vp = __shfl_xor(v, 1, 32);
      float sn, cs;
      __sincosf((float)(m & 511) * inv, &sn, &cs);
      qk[(size_t)m * 2048 + col] = odd ? (v * cs + vp * sn) : (v * cs - vp * sn);
    }
  }
}

// ---------------------------------------------------------------------------
// Kernel 2: logits[b,h,m,n] = sum_d q[b,m,h,d] * k[b,n,h,d], masked + scaled.
// One wave32 per 16x32 logits tile: two accumulators (N and N+16) sharing the
// Q fragment; 2x16 v_wmma_f32_16x16x4_f32 per wave (K = 64).
// q at qk[row][h*128 + d], k at qk[row][h*128 + 64 + d]; all fragment loads
// are aligned float2. Stores are lane-contiguous along N (coalesced).
// ---------------------------------------------------------------------------
__global__ __launch_bounds__(256) void gp_attn_logits(
    const float* __restrict__ qk, const int* __restrict__ mask,
    float* __restrict__ out)
{
  const int wave = blockIdx.x * 8 + ((int)threadIdx.x >> 5);
  const int lane = (int)threadIdx.x & 31;
  const int l15  = lane & 15;
  const int kh   = lane >> 4;
  const int bh   = wave >> 9;                  // 0..127 (32 M-tiles x 16 N-pairs)
  const int b    = bh >> 4;
  const int h    = bh & 15;
  const int tm   = ((wave >> 4) & 31) << 4;    // M tile base, S = 512
  const int tn0  = (wave & 15) << 5;           // N pair base
  const int tn1  = tn0 + 16;

  const float* qp  = qk + (size_t)(b * 512 + tm  + l15) * 2048 + h * 128 + (kh << 1);
  const float* kp0 = qk + (size_t)(b * 512 + tn0 + l15) * 2048 + h * 128 + 64 + (kh << 1);
  const float* kp1 = kp0 + (size_t)16 * 2048;

  v8f acc0 = {};
  v8f acc1 = {};
#pragma unroll
  for (int k0 = 0; k0 < 64; k0 += 4) {
    v2f a  = *(const v2f*)(qp  + k0);
    v2f b0 = *(const v2f*)(kp0 + k0);
    v2f b1 = *(const v2f*)(kp1 + k0);
    acc0 = __builtin_amdgcn_wmma_f32_16x16x4_f32(false, a, false, b0,
                                                 (short)0, acc0, false, false);
    acc1 = __builtin_amdgcn_wmma_f32_16x16x4_f32(false, a, false, b1,
                                                 (short)0, acc1, false, false);
  }

  const int   n0   = tn0 + l15;
  const int   n1   = tn1 + l15;
  const float pad0 = (float)mask[b * 512 + n0];
  const float pad1 = (float)mask[b * 512 + n1];
  const float np0  = (1.0f - pad0) * NEG_BIG;
  const float np1  = (1.0f - pad1) * NEG_BIG;
  float* obase = out + ((size_t)(b * 16 + h) * 512) * 512;
#pragma unroll
  for (int r = 0; r < 8; ++r) {
    const int m = tm + r + (kh << 3);
    float v0 = acc0[r] * pad0 - np0;
    float v1 = acc1[r] * pad1 - np1;
    v0 = (m > n0) ? (v0 - NEG_BIG) : v0;       // strict-lower causal mask
    v1 = (m > n1) ? (v1 - NEG_BIG) : v1;
    obase[(size_t)m * 512 + n0] = v0 * 0.125f; // 1/sqrt(64)
    obase[(size_t)m * 512 + n1] = v1 * 0.125f;
  }
}

extern "C" void kernel_launch(void* const* d_in, const int* in_sizes, int n_in,
                              void* d_out, int out_size, void* d_ws, size_t ws_size,
                              hipStream_t stream) {
  const float* X    = (const float*)d_in[0];   // (8,512,768) fp32
  const int*   msk  = (const int*)d_in[1];     // (8,512) int32
  const float* W    = (const float*)d_in[2];   // (768,2048) fp32
  const float* bias = (const float*)d_in[3];   // (2048,) fp32
  float* out = (float*)d_out;                  // (8,16,512,512) fp32
  float* qk  = (float*)d_ws;                   // 4096*2048 fp32 = 32 MB

  // 4096x2048 GEMM1: 128 M-pairs x 128 N-tiles = 16384 waves, 8 waves/block.
  gp_gemm_rope<<<2048, 256, 0, stream>>>(X, W, bias, qk);
  // 8*16 (b,h) x 32 M-tiles x 16 N-pairs = 65536 waves, 8 waves/block.
  gp_attn_logits<<<8192, 256, 0, stream>>>(qk, msk, out);
}